// Multiprototypes_7121055777423
// MI455X (gfx1250) — compile-verified
//
#include <hip/hip_runtime.h>
#include <cstdint>
#include <cmath>

#define ENT_TH   0.95f
#define DIST_EPS 1e-12f
#define FD    64                  // feature dim
#define NK    3                   // centers per class
#define BLOCK 128                 // 4 wave32 per block
#define TILE  128                 // samples per TDM tile
#define TILE_ELEMS (TILE * FD)    // 8192 f32 = 32 KB per buffer

typedef unsigned int u32;
typedef u32 u32x4 __attribute__((ext_vector_type(4)));
typedef int i32x4 __attribute__((ext_vector_type(4)));
typedef int i32x8 __attribute__((ext_vector_type(8)));

// ---------------------------------------------------------------------------
// zero the 3 double accumulators in workspace (must happen every launch)
// ---------------------------------------------------------------------------
__global__ void mp_init_accum(double* __restrict__ acc) {
    if (threadIdx.x < 3) acc[threadIdx.x] = 0.0;
}

// ---------------------------------------------------------------------------
// acc += ||xv - cv||^2 for one float4 chunk
// ---------------------------------------------------------------------------
__device__ __forceinline__ void accum_sq(const float4 xv, const float4 cv, float& acc) {
    float d;
    d = xv.x - cv.x; acc = fmaf(d, d, acc);
    d = xv.y - cv.y; acc = fmaf(d, d, acc);
    d = xv.z - cv.z; acc = fmaf(d, d, acc);
    d = xv.w - cv.w; acc = fmaf(d, d, acc);
}

// ---------------------------------------------------------------------------
// Tensor Data Mover: DMA one contiguous 1-D tile of f32 from global -> LDS.
// D# per CDNA5 ISA ch.8 (CDNA5 ISA): group0 = control/addresses,
// group1 = dims/strides, groups 2-3 zero (<=2D tensor).
// ---------------------------------------------------------------------------
__device__ __forceinline__ void tdm_load_tile(u32 lds_off,
                                              unsigned long long gaddr,
                                              u32 tile_elems,
                                              u32 tensor_elems) {
    u32x4 g0;
    g0.x = 1u;                                              // count=1, user desc
    g0.y = lds_off;                                         // lds_addr (bytes)
    g0.z = (u32)gaddr;                                      // global_addr[31:0]
    g0.w = (u32)((gaddr >> 32) & 0x01FFFFFFull)             // global_addr[56:32]
         | (2u << 30);                                      // type=2 ("image")
    i32x8 g1;
    g1[0] = (int)(2u << 16);                                // wg_mask=0, data_size=4B
    g1[1] = (int)((tensor_elems & 0xFFFFu) << 16);          // abar=0 | tensor_dim0.lo16
    g1[2] = (int)(((tensor_elems >> 16) & 0xFFFFu)          // tensor_dim0.hi16
         | (1u << 16));                                     // tensor_dim1 = 1
    g1[3] = (int)((tile_elems & 0xFFFFu) << 16);            // tile_dim0
    g1[4] = 1;                                              // tile_dim1=1, tile_dim2=0
    g1[5] = (int)tile_elems;                                // tensor_dim0_stride.lo32
    g1[6] = 0;                                              // stride hi / dim1_stride
    g1[7] = 0;
    i32x4 z4 = {0, 0, 0, 0};
#if __clang_major__ >= 23
    i32x8 z8 = {0, 0, 0, 0, 0, 0, 0, 0};
    __builtin_amdgcn_tensor_load_to_lds(g0, g1, z4, z4, z8, 0);   // 6-arg (clang-23)
#else
    __builtin_amdgcn_tensor_load_to_lds(g0, g1, z4, z4, 0);       // 5-arg (ROCm 7.2)
#endif
}

// ---------------------------------------------------------------------------
// main kernel: TDM double-buffered x tiles in LDS, per-sample distances +
// entropy gate, wave32/LDS reduction, f64 global atomics.
// acc[0]=sum(min_pos), acc[1]=sum(min_pos*mask), acc[2]=sum(mask)
// ---------------------------------------------------------------------------
__global__ __launch_bounds__(BLOCK) void mp_main(
    const float* __restrict__ x,
    const int*   __restrict__ labels,
    const float* __restrict__ centers,
    double*      __restrict__ acc,
    int n)
{
    __shared__ float sbuf[2][TILE_ELEMS];                   // 64 KB double buffer

    const int nTiles = (n + TILE - 1) / TILE;
    const int wid    = threadIdx.x >> 5;
    const int lane   = threadIdx.x & 31;
    const bool issuer = (wid == 0);                         // TENSORcnt is per-wave

    const unsigned long long xbase = (unsigned long long)(uintptr_t)x;
    const unsigned long long totElems = (unsigned long long)n * FD;

    float t_min = 0.0f, t_msk = 0.0f, t_cnt = 0.0f;

    int tile = (int)blockIdx.x;
    if (tile < nTiles && issuer) {
        const u32 lds0 = (u32)(uintptr_t)&sbuf[0][0];       // flat addr[31:0] = LDS off
        tdm_load_tile(lds0, xbase + (unsigned long long)tile * (TILE_ELEMS * 4ull),
                      TILE_ELEMS,
                      (u32)(totElems - (unsigned long long)tile * TILE_ELEMS));
    }

    int buf = 0;
    for (; tile < nTiles; tile += (int)gridDim.x, buf ^= 1) {
        const int nextTile = tile + (int)gridDim.x;
        if (issuer) {
            if (nextTile < nTiles) {
                const u32 lds1 = (u32)(uintptr_t)&sbuf[buf ^ 1][0];
                tdm_load_tile(lds1,
                              xbase + (unsigned long long)nextTile * (TILE_ELEMS * 4ull),
                              TILE_ELEMS,
                              (u32)(totElems - (unsigned long long)nextTile * TILE_ELEMS));
                __builtin_amdgcn_s_wait_tensorcnt(1);       // current tile complete
            } else {
                __builtin_amdgcn_s_wait_tensorcnt(0);       // drain
            }
        }
        __syncthreads();                                    // tile[buf] visible to all

        const int i = tile * TILE + (int)threadIdx.x;
        if (i < n) {
            if (threadIdx.x == 0 && nextTile < nTiles)
                __builtin_prefetch(labels + (size_t)nextTile * TILE, 0, 0);

            const float4* __restrict__ xr =
                (const float4*)&sbuf[buf][(size_t)threadIdx.x * FD];
            const int lab = labels[i];
            const float4* __restrict__ cv =
                (const float4*)(centers + (size_t)lab * (NK * FD));

            float a0 = 0.0f, a1 = 0.0f, a2 = 0.0f;
            const int rot = threadIdx.x & 15;               // spread LDS banks
            #pragma unroll
            for (int j = 0; j < FD / 4; ++j) {
                const int jj = (j + rot) & 15;
                const float4 xj = xr[jj];
                accum_sq(xj, cv[jj +  0], a0);
                accum_sq(xj, cv[jj + 16], a1);
                accum_sq(xj, cv[jj + 32], a2);
            }

            const float mn  = fminf(a0, fminf(a1, a2));
            const float mx  = fmaxf(a0, fmaxf(a1, a2));
            const float mid = (a0 + a1 + a2) - mn - mx;

            const float d0 = mn  + DIST_EPS;
            const float d1 = mid + DIST_EPS;
            const float p  = d0 / (d0 + d1);                // p <= 0.5
            const float q  = 1.0f - p;
            const float ent = -(p * log2f(p) + q * log2f(q));

            t_min += mn;
            if (ent <= ENT_TH) {                            // NaN -> false (matches jnp)
                t_msk += mn;
                t_cnt += 1.0f;
            }
        }
        __syncthreads();                                    // buffer reuse safe
    }

    // ---- wave32 reduction ----
    #pragma unroll
    for (int off = 16; off > 0; off >>= 1) {
        t_min += __shfl_down(t_min, off, 32);
        t_msk += __shfl_down(t_msk, off, 32);
        t_cnt += __shfl_down(t_cnt, off, 32);
    }

    // ---- cross-wave (4 waves) via LDS ----
    __shared__ float s0[BLOCK / 32];
    __shared__ float s1[BLOCK / 32];
    __shared__ float s2[BLOCK / 32];
    if (lane == 0) { s0[wid] = t_min; s1[wid] = t_msk; s2[wid] = t_cnt; }
    __syncthreads();

    if (wid == 0) {
        float v0 = (lane < BLOCK / 32) ? s0[lane] : 0.0f;
        float v1 = (lane < BLOCK / 32) ? s1[lane] : 0.0f;
        float v2 = (lane < BLOCK / 32) ? s2[lane] : 0.0f;
        #pragma unroll
        for (int off = 2; off > 0; off >>= 1) {
            v0 += __shfl_down(v0, off, 32);
            v1 += __shfl_down(v1, off, 32);
            v2 += __shfl_down(v2, off, 32);
        }
        if (lane == 0) {
            atomicAdd(acc + 0, (double)v0);                 // global_atomic_add_f64
            atomicAdd(acc + 1, (double)v1);
            atomicAdd(acc + 2, (double)v2);
        }
    }
}

// ---------------------------------------------------------------------------
// finalize: loss1 = sum_min / B ; loss2 = sum_masked / cnt
// ---------------------------------------------------------------------------
__global__ void mp_finalize(const double* __restrict__ acc,
                            float* __restrict__ out, int n) {
    if (threadIdx.x == 0 && blockIdx.x == 0) {
        out[0] = (float)(acc[0] / (double)n);
        out[1] = (float)(acc[1] / acc[2]);
    }
}

// ---------------------------------------------------------------------------
// launch
// ---------------------------------------------------------------------------
extern "C" void kernel_launch(void* const* d_in, const int* in_sizes, int n_in,
                              void* d_out, int out_size, void* d_ws, size_t ws_size,
                              hipStream_t stream) {
    const float* x       = (const float*)d_in[0];
    const int*   labels  = (const int*)  d_in[1];
    const float* centers = (const float*)d_in[2];
    const int    n       = in_sizes[1];           // B (labels count)

    double* acc = (double*)d_ws;                  // 3 doubles of scratch
    float*  out = (float*)d_out;                  // 2 fp32 outputs

    mp_init_accum<<<1, 64, 0, stream>>>(acc);

    const int nTiles = (n + TILE - 1) / TILE;
    const int blocks = nTiles < 1024 ? nTiles : 1024;
    mp_main<<<blocks, BLOCK, 0, stream>>>(x, labels, centers, acc, n);

    mp_finalize<<<1, 64, 0, stream>>>(acc, out, n);
}